// Link_54889682043184
// MI455X (gfx1250) — compile-verified
//
#include <hip/hip_runtime.h>

// Problem dims (compile-time constants from the reference)
#define BB 2
#define LL 512
#define TT 512
#define HH 768
#define HC 32   // H-columns per scan workgroup (LDS slab = 512*32*4 = 64KB)

typedef float v2f __attribute__((ext_vector_type(2)));
typedef float v8f __attribute__((ext_vector_type(8)));
typedef int   v4i __attribute__((ext_vector_type(4)));
typedef int   v8i __attribute__((ext_vector_type(8)));

// ---------------------------------------------------------------------------
// Kernel 1: per (b,t) find LAST row l with tag_to_token>0; also zero counters.
// ---------------------------------------------------------------------------
__global__ void k_lastrow_zero(const float* __restrict__ t2t,
                               int* __restrict__ lastrow,
                               float* __restrict__ cnt) {
    int tid = blockIdx.x * blockDim.x + threadIdx.x;
    if (tid < BB * LL) cnt[tid] = 0.0f;
    if (tid >= BB * TT) return;
    int b = tid / TT;
    int t = tid - b * TT;
    const float* p = t2t + (size_t)b * LL * TT + t;
    int last = -1;
    for (int l = 0; l < LL; ++l) {
        if (p[(size_t)l * TT] > 0.0f) last = l;   // coalesced across t
    }
    lastrow[tid] = last;
}

// ---------------------------------------------------------------------------
// Kernel 2: per-row survivor counts (row-normalization denominators).
// ---------------------------------------------------------------------------
__global__ void k_count(const int* __restrict__ lastrow, float* __restrict__ cnt) {
    int tid = blockIdx.x * blockDim.x + threadIdx.x;
    if (tid >= BB * TT) return;
    int b = tid / TT;
    int last = lastrow[tid];
    if (last >= 0) atomicAdd(&cnt[b * LL + last], 1.0f);
}

// ---------------------------------------------------------------------------
// Kernel 3: materialize normalized t2t matrix (input A of the WMMA GEMM).
// ---------------------------------------------------------------------------
__global__ void k_build_t2t(const int* __restrict__ lastrow,
                            const float* __restrict__ cnt,
                            float* __restrict__ t2tn) {
    int idx = blockIdx.x * 256 + threadIdx.x;      // < BB*LL*TT = 524288
    int t = idx & (TT - 1);
    int l = (idx >> 9) & (LL - 1);
    int b = idx >> 18;
    float v = 0.0f;
    if (lastrow[b * TT + t] == l) {
        float s = cnt[b * LL + l];
        v = 1.0f / (s > 0.0f ? s : 1.0f);
    }
    t2tn[idx] = v;
}

// ---------------------------------------------------------------------------
// Kernel 4: deduce_child. One 128-thread block per (b,i); candidate vector c
// lives in LDS. Sequential sweep j=i+1..L-1; branch on c[j] is uniform.
// ---------------------------------------------------------------------------
__global__ void k_child(const float* __restrict__ gm, float* __restrict__ child_out) {
    __shared__ float c_s[LL];
    __shared__ float red[128];
    int i   = blockIdx.x;
    int b   = blockIdx.y;
    int tid = threadIdx.x;                // 0..127, each owns 4 elements
    int t0  = tid * 4;
    const float* gmb = gm + (size_t)b * LL * LL;

    *(float4*)&c_s[t0] = *(const float4*)(gmb + (size_t)i * LL + t0);
    __syncthreads();

    for (int j = i + 1; j < LL; ++j) {
        float cj = c_s[j];                // uniform broadcast read
        __syncthreads();                  // reads complete before writes
        if (cj > 0.0f) {
            float4 g4 = *(const float4*)(gmb + (size_t)j * LL + t0);
            float gv[4] = {g4.x, g4.y, g4.z, g4.w};
            int d = j - t0;
            if (d >= 0 && d < 4) gv[d] = 0.0f;     // diagonal zeroed in row j
            float4 c4 = *(float4*)&c_s[t0];
            float cv[4] = {c4.x, c4.y, c4.z, c4.w};
            #pragma unroll
            for (int k = 0; k < 4; ++k)
                cv[k] = (cv[k] - cj * gv[k] > 0.0f) ? 1.0f : 0.0f;
            *(float4*)&c_s[t0] = make_float4(cv[0], cv[1], cv[2], cv[3]);
        }
        __syncthreads();                  // writes visible before next read
    }

    if (tid == 0) c_s[i] = 1.0f;
    __syncthreads();

    float4 c4 = *(float4*)&c_s[t0];
    red[tid] = c4.x + c4.y + c4.z + c4.w;
    __syncthreads();
    for (int s = 64; s > 0; s >>= 1) {
        if (tid < s) red[tid] += red[tid + s];
        __syncthreads();
    }
    float inv = 1.0f / red[0];
    float* dst = child_out + (size_t)b * LL * LL + (size_t)i * LL + t0;
    dst[0] = c4.x * inv; dst[1] = c4.y * inv; dst[2] = c4.z * inv; dst[3] = c4.w * inv;
}

// ---------------------------------------------------------------------------
// Kernel 5: out0 = t2tn (LxT) @ inputs (TxH) per batch, via f32 WMMA 16x16x4.
// Workgroup = 8 waves covering a 128(M) x 16(N) C stripe. The shared B panel
// (512 x 16 f32 = 32KB) is DMA'd into LDS once by the Tensor Data Mover
// (tensor_load_to_lds, TENSORcnt), then all waves feed B fragments from LDS
// while streaming their private A fragments from global.
//
// NOTE: the LDS panel's address is passed into the asm as an (unused) VGPR
// operand so it escapes; combined with the "memory" clobber this tells the
// optimizer the TDM writes the array. ("m" constraints crash the AMDGPU
// inline-asm selector; a plain register operand is the supported way.)
//
// Fragment layouts (ISA 7.12.2, 32-bit):
//   A 16x4 : M=lane&15, K = k0 + 2*(lane>>4) + vgpr
//   B 4x16 : N=lane&15, K = k0 + 2*(lane>>4) + vgpr
//   C 16x16: N=lane&15, M = vgpr + 8*(lane>>4)
// ---------------------------------------------------------------------------
__global__ void k_gemm_wmma(const float* __restrict__ A,
                            const float* __restrict__ Bm,
                            float* __restrict__ C) {
    __shared__ float ldsB[TT * 16];       // 512 rows x 16 cols = 32KB, LDS offset 0

    int b    = blockIdx.z;
    int n0   = blockIdx.x * 16;
    int m0   = blockIdx.y * 128 + (threadIdx.x >> 5) * 16;
    int lane = threadIdx.x & 31;

    const float* Ab = A  + (size_t)b * LL * TT;
    const float* Bb = Bm + (size_t)b * TT * HH;
    float*       Cb = C  + (size_t)b * LL * HH;

    // --- TDM: async-copy the 512x16 B panel into LDS (wave 0 issues) -------
    if (threadIdx.x < 32) {
        unsigned long long gaddr = (unsigned long long)(const void*)(Bb + n0);
        v4i g0;
        g0[0] = 1;                                   // count=1, user descriptor
        g0[1] = 0;                                   // lds_addr = 0 (ldsB base)
        g0[2] = (int)(gaddr & 0xFFFFFFFFull);        // global_addr[31:0]
        g0[3] = (int)((gaddr >> 32) & 0x01FFFFFFull) // global_addr[56:32]
              | (int)0x80000000;                     // type=2 (image)
        v8i g1;
        g1[0] = 0x00020000;                          // data_size=4B; no pad/iter
        g1[1] = (HH & 0xFFFF) << 16;                 // tensor_dim0 = H (lo16)
        g1[2] = ((HH >> 16) & 0xFFFF) | ((TT & 0xFFFF) << 16); // dim0 hi | dim1 lo
        g1[3] = ((TT >> 16) & 0xFFFF) | (16 << 16);  // dim1 hi | tile_dim0 = 16
        g1[4] = TT;                                  // tile_dim1 = 512, tile_dim2 = 0
        g1[5] = HH;                                  // tensor_dim0_stride[31:0] = H
        g1[6] = 0;                                   // stride hi | dim1_stride lo
        g1[7] = 0;
        float* lds_escape = &ldsB[0];                // escapes -> array kept/written
        asm volatile("tensor_load_to_lds %0, %1"
                     :: "s"(g0), "s"(g1), "v"(lds_escape)
                     : "memory");
        __builtin_amdgcn_s_wait_tensorcnt(0);
    }
    __syncthreads();

    int row = lane & 15;                  // M (A) / N (B) index within tile
    int g   = lane >> 4;                  // lane-half selects K pair

    v8f acc = {};
    #pragma unroll 4
    for (int k0 = 0; k0 < TT; k0 += 4) {
        const float* ap = Ab + (size_t)(m0 + row) * TT + (k0 + 2 * g);
        __builtin_prefetch(ap + 32, 0, 1);           // global_prefetch_b8
        v2f av, bv;
        av.x = ap[0];  av.y = ap[1];                 // contiguous pair (b64)
        bv.x = ldsB[(k0 + 2 * g) * 16 + row];        // ds_load, conflict-free
        bv.y = ldsB[(k0 + 2 * g + 1) * 16 + row];
        acc = __builtin_amdgcn_wmma_f32_16x16x4_f32(
            /*neg_a=*/false, av, /*neg_b=*/false, bv,
            /*c_mod=*/(short)0, acc, /*reuse_a=*/false, /*reuse_b=*/false);
    }

    #pragma unroll
    for (int rr = 0; rr < 8; ++rr)
        Cb[(size_t)(m0 + rr + 8 * g) * HH + n0 + row] = acc[rr];
}

// ---------------------------------------------------------------------------
// Kernel 6: sequential scan over rows i=511..0 on an H-slab held in LDS,
// then add seq_table / depth_table rows and write final output.
// ---------------------------------------------------------------------------
__global__ void k_scan(float* __restrict__ out, const float* __restrict__ child,
                       const int* __restrict__ tag_depth,
                       const float* __restrict__ seq_table,
                       const float* __restrict__ depth_table) {
    extern __shared__ float s[];          // LL*HC floats = 64KB
    __shared__ float red[256];
    int b  = blockIdx.y;
    int h0 = blockIdx.x * HC;
    int tid = threadIdx.x;                // 256 threads
    float* outb = out + (size_t)b * LL * HH;
    const float* chb = child + (size_t)b * LL * LL;

    for (int idx = tid; idx < LL * HC; idx += 256) {
        int i = idx >> 5, h = idx & (HC - 1);
        s[idx] = outb[(size_t)i * HH + h0 + h];
    }
    __syncthreads();

    int h    = tid & (HC - 1);            // lane = h -> wave-uniform child loads
    int part = tid >> 5;                  // 8 partial sums over l
    for (int i = LL - 1; i >= 0; --i) {
        const float* crow = chb + (size_t)i * LL;
        float acc = 0.0f;
        int l0 = part * 64;
        #pragma unroll 4
        for (int l = l0; l < l0 + 64; ++l)
            acc += crow[l] * s[l * HC + h];
        red[tid] = acc;
        __syncthreads();
        if (part == 0) {
            float tot = red[h] + red[32 + h] + red[64 + h] + red[96 + h]
                      + red[128 + h] + red[160 + h] + red[192 + h] + red[224 + h];
            s[i * HC + h] = tot;          // replace row i (rows >i already new)
        }
        __syncthreads();
    }

    for (int idx = tid; idx < LL * HC; idx += 256) {
        int i = idx >> 5, hh = idx & (HC - 1);
        int d = tag_depth[b * LL + i];
        float v = s[idx]
                + seq_table[(size_t)i * HH + h0 + hh]
                + depth_table[(size_t)d * HH + h0 + hh];
        outb[(size_t)i * HH + h0 + hh] = v;
    }
}

// ---------------------------------------------------------------------------
extern "C" void kernel_launch(void* const* d_in, const int* in_sizes, int n_in,
                              void* d_out, int out_size, void* d_ws, size_t ws_size,
                              hipStream_t stream) {
    const float* inputs  = (const float*)d_in[0];   // (B,T,H)
    const float* t2t     = (const float*)d_in[1];   // (B,L,T)
    const float* gm      = (const float*)d_in[2];   // (B,L,L)
    const int*   tdepth  = (const int*)  d_in[3];   // (B,L)
    const float* seq     = (const float*)d_in[4];   // (512,H)
    const float* dtab    = (const float*)d_in[5];   // (64,H)

    float* out   = (float*)d_out;                       // (B,L,H)
    float* child = out + (size_t)BB * LL * HH;          // (B,L,L), 2nd output

    char*  ws      = (char*)d_ws;
    int*   lastrow = (int*)  ws;                        // B*T ints
    float* cnt     = (float*)(ws + 4096);               // B*L floats
    float* t2tn    = (float*)(ws + 8192);               // B*L*T floats (2MB)

    k_lastrow_zero<<<(BB * TT + 255) / 256, 256, 0, stream>>>(t2t, lastrow, cnt);
    k_count       <<<(BB * TT + 255) / 256, 256, 0, stream>>>(lastrow, cnt);
    k_build_t2t   <<<(BB * LL * TT) / 256, 256, 0, stream>>>(lastrow, cnt, t2tn);
    k_child       <<<dim3(LL, BB), 128, 0, stream>>>(gm, child);

    // 8 waves/block, each wave one 16x16 tile stacked along M, shared B panel
    k_gemm_wmma<<<dim3(HH / 16, LL / 128, BB), 256, 0, stream>>>(t2tn, inputs, out);

    size_t smem = (size_t)LL * HC * sizeof(float);      // 64KB slab
    k_scan<<<dim3(HH / HC, BB), 256, smem, stream>>>(out, child, tdepth, seq, dtab);
}